// expN1_1468878815467
// MI455X (gfx1250) — compile-verified
//
#include <hip/hip_runtime.h>

// out[t,n] = 1 - n4*x[t-4]        (elementwise, delays along T, zeros pre-history)
//   n1 = 1 - x[t]  *c1[t]
//   n2 = 1 - n1*c2[t]*x[t-1]
//   n3 = 1 - n2*c3[t]*x[t-2]
//   n4 = 1 - n3*c4[t]*x[t-3]
//
// Memory-bound (~1.75 flop/byte): stream x once, write out once, 128-bit
// non-temporal accesses, 5-deep register sliding window for the delay taps.

typedef float v4f __attribute__((ext_vector_type(4)));

#define TCHUNK 32  // T-rows per block-chunk; 4-row halo => ~6% extra traffic

__global__ __launch_bounds__(256) void sc_poly_v4(
    const float* __restrict__ x,
    const float* __restrict__ c1, const float* __restrict__ c2,
    const float* __restrict__ c3, const float* __restrict__ c4,
    float* __restrict__ out, int NV /* N/4 */, int T)
{
    const int col = blockIdx.x * blockDim.x + threadIdx.x;   // vec4 column
    if (col >= NV) return;
    const int t0 = blockIdx.y * TCHUNK;
    if (t0 >= T) return;
    int tend = t0 + TCHUNK; if (tend > T) tend = T;

    const v4f* __restrict__ xv = (const v4f*)x;
    v4f* __restrict__ ov = (v4f*)out;

    const v4f z = {0.f, 0.f, 0.f, 0.f};
    // 4-row history halo (zeros before t=0), kept in registers
    v4f xm1 = (t0 >= 1) ? __builtin_nontemporal_load(&xv[(size_t)(t0 - 1) * NV + col]) : z;
    v4f xm2 = (t0 >= 2) ? __builtin_nontemporal_load(&xv[(size_t)(t0 - 2) * NV + col]) : z;
    v4f xm3 = (t0 >= 3) ? __builtin_nontemporal_load(&xv[(size_t)(t0 - 3) * NV + col]) : z;
    v4f xm4 = (t0 >= 4) ? __builtin_nontemporal_load(&xv[(size_t)(t0 - 4) * NV + col]) : z;

#pragma unroll 8
    for (int t = t0; t < tend; ++t) {
        v4f xt = __builtin_nontemporal_load(&xv[(size_t)t * NV + col]);
        // wave-uniform coefficients -> scalar loads (s_load_b32)
        const float a1 = c1[t], a2 = c2[t], a3 = c3[t], a4 = c4[t];

        v4f n1 = 1.0f - xt * a1;
        v4f n2 = 1.0f - n1 * (a2 * xm1);
        v4f n3 = 1.0f - n2 * (a3 * xm2);
        v4f n4 = 1.0f - n3 * (a4 * xm3);
        v4f o  = 1.0f - n4 * xm4;

        __builtin_nontemporal_store(o, &ov[(size_t)t * NV + col]);

        xm4 = xm3; xm3 = xm2; xm2 = xm1; xm1 = xt;  // slide window (renamed regs)
    }
}

// Scalar fallback for N not divisible by 4 (not hit for the given shapes).
__global__ __launch_bounds__(256) void sc_poly_s(
    const float* __restrict__ x,
    const float* __restrict__ c1, const float* __restrict__ c2,
    const float* __restrict__ c3, const float* __restrict__ c4,
    float* __restrict__ out, int N, int T)
{
    const int col = blockIdx.x * blockDim.x + threadIdx.x;
    if (col >= N) return;
    const int t0 = blockIdx.y * TCHUNK;
    if (t0 >= T) return;
    int tend = t0 + TCHUNK; if (tend > T) tend = T;

    float xm1 = (t0 >= 1) ? x[(size_t)(t0 - 1) * N + col] : 0.f;
    float xm2 = (t0 >= 2) ? x[(size_t)(t0 - 2) * N + col] : 0.f;
    float xm3 = (t0 >= 3) ? x[(size_t)(t0 - 3) * N + col] : 0.f;
    float xm4 = (t0 >= 4) ? x[(size_t)(t0 - 4) * N + col] : 0.f;

#pragma unroll 4
    for (int t = t0; t < tend; ++t) {
        float xt = x[(size_t)t * N + col];
        const float a1 = c1[t], a2 = c2[t], a3 = c3[t], a4 = c4[t];
        float n1 = 1.0f - xt * a1;
        float n2 = 1.0f - n1 * (a2 * xm1);
        float n3 = 1.0f - n2 * (a3 * xm2);
        float n4 = 1.0f - n3 * (a4 * xm3);
        out[(size_t)t * N + col] = 1.0f - n4 * xm4;
        xm4 = xm3; xm3 = xm2; xm2 = xm1; xm1 = xt;
    }
}

extern "C" void kernel_launch(void* const* d_in, const int* in_sizes, int n_in,
                              void* d_out, int out_size, void* d_ws, size_t ws_size,
                              hipStream_t stream)
{
    const float* x  = (const float*)d_in[0];
    const float* c1 = (const float*)d_in[1];
    const float* c2 = (const float*)d_in[2];
    const float* c3 = (const float*)d_in[3];
    const float* c4 = (const float*)d_in[4];
    float* out = (float*)d_out;

    const int T = in_sizes[1];                       // 256 (length of c1)
    const long long total = (long long)in_sizes[0];  // T * N
    const int N = (int)(total / T);                  // 131072

    const int nchunks = (T + TCHUNK - 1) / TCHUNK;   // 8

    if ((N & 3) == 0) {
        const int NV = N >> 2;                       // 32768 float4 columns
        dim3 grid((NV + 255) / 256, nchunks);        // 128 x 8 blocks, 256 thr (8 waves)
        sc_poly_v4<<<grid, dim3(256), 0, stream>>>(x, c1, c2, c3, c4, out, NV, T);
    } else {
        dim3 grid((N + 255) / 256, nchunks);
        sc_poly_s<<<grid, dim3(256), 0, stream>>>(x, c1, c2, c3, c4, out, N, T);
    }
}